// Net1_16793322127388
// MI455X (gfx1250) — compile-verified
//
#include <hip/hip_runtime.h>
#include <hip/hip_bf16.h>

typedef __attribute__((ext_vector_type(2))) float v2f;
typedef __attribute__((ext_vector_type(8))) float v8f;

#define FIN 128
#define DIM 32
#define CLS 10
#define FTOT (FIN + 3 * DIM)   // 224

// ---------------------------------------------------------------------------
// C = H(NxK) @ W(Kx32) [+ bias], fp32 WMMA 16x16x4.
// One wave computes a 32-row x 32-col output block: two 16-row A tiles
// sharing the same B fragments -> 4 WMMAs per 6 loads per k-step.
// Out-of-range rows are CLAMPED (not masked): a bogus A row only corrupts
// the matching D row, which the guarded store never writes. This keeps the
// inner loop 100% branchless (no exec-mask save/restore).
//
// A layout: lanes 0-15 hold (M=lane, K=k,k+1), lanes 16-31 hold K=k+2,k+3.
// B layout: lanes 0-15 hold (N=lane, K=k,k+1), lanes 16-31 hold K=k+2,k+3.
// D layout: VGPR r -> row r (lanes 0-15) / row r+8 (lanes 16-31), col lane&15.
// ---------------------------------------------------------------------------
__global__ void gemm_n32_wmma(const float* __restrict__ H,
                              const float* __restrict__ W,
                              const float* __restrict__ bias,
                              float* __restrict__ out, int N, int K) {
    const int lane = threadIdx.x & 31;
    const int wid  = threadIdx.x >> 5;
    const int blk  = blockIdx.x * (blockDim.x >> 5) + wid;   // 32-row block
    const int row0 = blk * 32;
    if (row0 >= N) return;                                   // wave-uniform

    const int m  = lane & 15;          // row-in-tile for A, col for B/D
    const int kh = (lane >> 4) * 2;    // K sub-offset (0 or 2)
    const int rowA = min(row0 + m,      N - 1);   // clamped, always valid
    const int rowB = min(row0 + 16 + m, N - 1);

    const float* __restrict__ pA = H + (size_t)rowA * K + kh;
    const float* __restrict__ pB = H + (size_t)rowB * K + kh;

    v8f c00 = {}, c01 = {}, c10 = {}, c11 = {};

    for (int k = 0; k < K; k += 4) {
        const int kk = k + kh;
        const v2f a0 = *(const v2f*)(pA + k);
        const v2f a1 = *(const v2f*)(pB + k);
        v2f b0, b1;
        b0.x = W[(size_t)kk * DIM + m];
        b0.y = W[(size_t)(kk + 1) * DIM + m];
        b1.x = W[(size_t)kk * DIM + m + 16];
        b1.y = W[(size_t)(kk + 1) * DIM + m + 16];

        c00 = __builtin_amdgcn_wmma_f32_16x16x4_f32(false, a0, false, b0,
                                                    (short)0, c00, false, false);
        c01 = __builtin_amdgcn_wmma_f32_16x16x4_f32(false, a0, false, b1,
                                                    (short)0, c01, false, false);
        c10 = __builtin_amdgcn_wmma_f32_16x16x4_f32(false, a1, false, b0,
                                                    (short)0, c10, false, false);
        c11 = __builtin_amdgcn_wmma_f32_16x16x4_f32(false, a1, false, b1,
                                                    (short)0, c11, false, false);
    }

    const int rbase = (lane >> 4) * 8;
    const float bv0 = bias ? bias[m]      : 0.0f;
    const float bv1 = bias ? bias[m + 16] : 0.0f;
#pragma unroll
    for (int r = 0; r < 8; ++r) {
        const int orow0 = row0 + rbase + r;        // tile 0
        const int orow1 = orow0 + 16;              // tile 1
        if (orow0 < N) {
            out[(size_t)orow0 * DIM + m]      = c00[r] + bv0;
            out[(size_t)orow0 * DIM + m + 16] = c01[r] + bv1;
        }
        if (orow1 < N) {
            out[(size_t)orow1 * DIM + m]      = c10[r] + bv0;
            out[(size_t)orow1 * DIM + m + 16] = c11[r] + bv1;
        }
    }
}

// ---------------------------------------------------------------------------
// out[dst[e]*32 + f] += msg[src[e]*32 + f].  One wave per edge: coalesced
// 128B gather, 32 consecutive f32 atomics; src/dst wave-uniform.
// ---------------------------------------------------------------------------
__global__ void scatter_add(const float* __restrict__ msg,
                            const int* __restrict__ src,
                            const int* __restrict__ dst,
                            float* __restrict__ out, int E) {
    const int idx = blockIdx.x * blockDim.x + threadIdx.x;
    const int e = idx >> 5;
    if (e >= E) return;
    const int f = idx & 31;
    const int s = src[e];
    const int d = dst[e];
    atomicAdd(&out[(size_t)d * DIM + f], msg[(size_t)s * DIM + f]);
}

__global__ void relu_k(float* __restrict__ h, int n) {
    const int i = blockIdx.x * blockDim.x + threadIdx.x;
    if (i < n) h[i] = fmaxf(h[i], 0.0f);
}

// ---------------------------------------------------------------------------
// logits = [x|h1|h2|h3] @ fc_w + fc_b ; out = log_softmax(logits).
// One wave per node; fc_w staged in LDS; wave32 shfl_xor reductions.
// ---------------------------------------------------------------------------
__global__ void fc_logsoftmax(const float* __restrict__ x,
                              const float* __restrict__ h1,
                              const float* __restrict__ h2,
                              const float* __restrict__ h3,
                              const float* __restrict__ fc_w,
                              const float* __restrict__ fc_b,
                              float* __restrict__ out, int N) {
    __shared__ float sW[FTOT * CLS];
    __shared__ float sB[CLS];
    for (int i = threadIdx.x; i < FTOT * CLS; i += blockDim.x) sW[i] = fc_w[i];
    if (threadIdx.x < CLS) sB[threadIdx.x] = fc_b[threadIdx.x];
    __syncthreads();

    const int lane = threadIdx.x & 31;
    const int wid  = threadIdx.x >> 5;
    const int node = blockIdx.x * (blockDim.x >> 5) + wid;
    if (node >= N) return;

    float acc[CLS];
#pragma unroll
    for (int c = 0; c < CLS; ++c) acc[c] = 0.0f;

    const float* xr = x + (size_t)node * FIN;
#pragma unroll
    for (int t = 0; t < FIN / 32; ++t) {
        const int k = t * 32 + lane;
        const float v = xr[k];
        const float* wr = &sW[k * CLS];
#pragma unroll
        for (int c = 0; c < CLS; ++c) acc[c] = fmaf(v, wr[c], acc[c]);
    }
    const float* hs[3] = {h1, h2, h3};
#pragma unroll
    for (int j = 0; j < 3; ++j) {
        const int k = FIN + j * DIM + lane;
        const float v = hs[j][(size_t)node * DIM + lane];
        const float* wr = &sW[k * CLS];
#pragma unroll
        for (int c = 0; c < CLS; ++c) acc[c] = fmaf(v, wr[c], acc[c]);
    }

    // butterfly reduce across the wave32 -> every lane holds full dot products
#pragma unroll
    for (int c = 0; c < CLS; ++c) {
        float v = acc[c];
#pragma unroll
        for (int off = 16; off > 0; off >>= 1) v += __shfl_xor(v, off, 32);
        acc[c] = v + sB[c];
    }

    float m = acc[0];
#pragma unroll
    for (int c = 1; c < CLS; ++c) m = fmaxf(m, acc[c]);
    float s = 0.0f;
#pragma unroll
    for (int c = 0; c < CLS; ++c) s += __expf(acc[c] - m);
    const float lse = m + __logf(s);

    if (lane == 0) {
#pragma unroll
        for (int c = 0; c < CLS; ++c)
            out[(size_t)node * CLS + c] = acc[c] - lse;
    }
}

// ---------------------------------------------------------------------------
extern "C" void kernel_launch(void* const* d_in, const int* in_sizes, int n_in,
                              void* d_out, int out_size, void* d_ws, size_t ws_size,
                              hipStream_t stream) {
    const float* x   = (const float*)d_in[0];
    const int*   src = (const int*)d_in[1];
    const int*   dst = (const int*)d_in[2];
    const float* Wn[3] = {(const float*)d_in[3], (const float*)d_in[6], (const float*)d_in[9]};
    const float* bn[3] = {(const float*)d_in[4], (const float*)d_in[7], (const float*)d_in[10]};
    const float* Ws[3] = {(const float*)d_in[5], (const float*)d_in[8], (const float*)d_in[11]};
    const float* fcw = (const float*)d_in[12];
    const float* fcb = (const float*)d_in[13];
    float* out = (float*)d_out;

    const int N = in_sizes[0] / FIN;
    const int E = in_sizes[1];

    // scratch layout: h1 | h2 | h3 | msg  (each N*32 f32)
    float* h1  = (float*)d_ws;
    float* h2  = h1 + (size_t)N * DIM;
    float* h3  = h2 + (size_t)N * DIM;
    float* msg = h3 + (size_t)N * DIM;

    const float* hin[3]  = {x, h1, h2};
    float*       hout[3] = {h1, h2, h3};
    const int    din[3]  = {FIN, DIM, DIM};

    const int gemmBlock = 256;                       // 8 waves -> 256 rows/block
    const int rowsPerBlock = (gemmBlock / 32) * 32;  // 256
    const int gemmGrid = (N + rowsPerBlock - 1) / rowsPerBlock;

    const long long eThreads = (long long)E * 32;
    const int scatGrid = (int)((eThreads + 255) / 256);
    const int reluGrid = (N * DIM + 255) / 256;

    for (int i = 0; i < 3; ++i) {
        // msg = h @ Wn
        gemm_n32_wmma<<<gemmGrid, gemmBlock, 0, stream>>>(hin[i], Wn[i], nullptr,
                                                          msg, N, din[i]);
        // hout = h @ Ws + bn   (accumulation target for the scatter-add)
        gemm_n32_wmma<<<gemmGrid, gemmBlock, 0, stream>>>(hin[i], Ws[i], bn[i],
                                                          hout[i], N, din[i]);
        // hout[dst] += msg[src]
        scatter_add<<<scatGrid, 256, 0, stream>>>(msg, src, dst, hout[i], E);
        // hout = relu(hout)
        relu_k<<<reluGrid, 256, 0, stream>>>(hout[i], N * DIM);
    }

    const int fcBlock = 256;                  // 8 waves -> 8 nodes per block
    const int fcGrid = (N + 7) / 8;
    fc_logsoftmax<<<fcGrid, fcBlock, 0, stream>>>(x, h1, h2, h3, fcw, fcb, out, N);
}